// Graph2VecSet2Set_40948218200428
// MI455X (gfx1250) — compile-verified
//
#include <hip/hip_runtime.h>

// ---------------- problem constants (fixed by the reference) ----------------
#define NN   32768
#define EE   262144
#define GG   256
#define DIN  256
#define DH   512
#define DOUT 512
#define STEPS 3

typedef float v2f __attribute__((ext_vector_type(2)));
typedef float v8f __attribute__((ext_vector_type(8)));

// ---------------- small helpers ----------------
__device__ __forceinline__ float sigmoidf_(float x) { return 1.0f / (1.0f + expf(-x)); }

// order-preserving float<->uint encoding for atomicMax on floats
__device__ __forceinline__ unsigned flt_flip(float f) {
  unsigned u = __float_as_uint(f);
  return (u & 0x80000000u) ? ~u : (u | 0x80000000u);
}
__device__ __forceinline__ float flt_unflip(unsigned u) {
  return (u & 0x80000000u) ? __uint_as_float(u & 0x7FFFFFFFu) : __uint_as_float(~u);
}

// ---------------- utility kernels ----------------
__global__ void zero_f32(float* __restrict__ p, int n) {
  int i = blockIdx.x * blockDim.x + threadIdx.x;
  if (i < n) p[i] = 0.0f;
}
__global__ void set_ones(float* __restrict__ p, int n) {
  int i = blockIdx.x * blockDim.x + threadIdx.x;
  if (i < n) p[i] = 1.0f;
}
__global__ void zero_small(unsigned* __restrict__ emax, float* __restrict__ denom, int g) {
  int i = blockIdx.x * blockDim.x + threadIdx.x;
  if (i < g) { emax[i] = 0u; denom[i] = 0.0f; }
}

// ---------------- degree / norm ----------------
__global__ void deg_edges(const long long* __restrict__ dst, float* __restrict__ deg, int e) {
  int i = blockIdx.x * blockDim.x + threadIdx.x;
  if (i < e) atomicAdd(&deg[(int)dst[i]], 1.0f);
}
__global__ void rsqrt_inplace(float* __restrict__ p, int n) {
  int i = blockIdx.x * blockDim.x + threadIdx.x;
  if (i < n) p[i] = rsqrtf(p[i]);
}
__global__ void norm_kernel(const long long* __restrict__ src, const long long* __restrict__ dst,
                            const float* __restrict__ dinv, float* __restrict__ nrm, int e) {
  int i = blockIdx.x * blockDim.x + threadIdx.x;
  if (i < e) nrm[i] = dinv[(int)src[i]] * dinv[(int)dst[i]];
}

// ============ LDS-staged WMMA fp32 GEMM: C[M,Nc] = A[M,K] @ B[K,Nc] ============
// Block = 256 threads = 8 waves -> 128x64 output tile (wave: 16 rows x 64 cols,
// 4 v8f accumulators). B panel [16 x 64] staged once per block in LDS, stored
// fragment-major Bs[col][k] so each lane's (k,k+1) B pair is contiguous.
#define KB 16
__global__ void wmma_gemm_f32_lds(const float* __restrict__ A, const float* __restrict__ B,
                                  float* __restrict__ C, int K, int Nc) {
  __shared__ float Bs[64][KB + 1];   // +1 pad: conflict-free lane access

  const int tid    = threadIdx.x;
  const int waveId = tid >> 5;
  const int lane   = tid & 31;
  const int half   = lane >> 4;      // 0: lanes 0-15, 1: lanes 16-31
  const int lr     = lane & 15;

  const int colBlocks = Nc >> 6;     // Nc / 64
  const int br  = blockIdx.x / colBlocks;
  const int bc  = blockIdx.x % colBlocks;
  const int tr  = (br << 7) + (waveId << 4);   // wave's 16 rows
  const int tc0 = bc << 6;                     // block's 64 cols

  // B staging assignment: thread -> (k row bkk, 4 cols starting at bcc)
  const int bkk = tid >> 4;          // 0..15
  const int bcc = (tid & 15) << 2;   // 0..60 step 4

  v8f acc0 = {}, acc1 = {}, acc2 = {}, acc3 = {};
  const float* Arow = A + (size_t)(tr + lr) * K;

  for (int k0 = 0; k0 < K; k0 += KB) {
    const float4 bv = *(const float4*)&B[(size_t)(k0 + bkk) * Nc + tc0 + bcc];
    __syncthreads();                 // previous chunk's Bs reads complete
    Bs[bcc + 0][bkk] = bv.x;
    Bs[bcc + 1][bkk] = bv.y;
    Bs[bcc + 2][bkk] = bv.z;
    Bs[bcc + 3][bkk] = bv.w;
    __syncthreads();
#pragma unroll
    for (int kk = 0; kk < KB; kk += 4) {
      const int ka = kk + (half << 1);         // k = vgpr + 2*half
      v2f a; a.x = Arow[k0 + ka]; a.y = Arow[k0 + ka + 1];
      v2f b0, b1, b2, b3;
      b0.x = Bs[lr     ][ka]; b0.y = Bs[lr     ][ka + 1];
      b1.x = Bs[lr + 16][ka]; b1.y = Bs[lr + 16][ka + 1];
      b2.x = Bs[lr + 32][ka]; b2.y = Bs[lr + 32][ka + 1];
      b3.x = Bs[lr + 48][ka]; b3.y = Bs[lr + 48][ka + 1];
      acc0 = __builtin_amdgcn_wmma_f32_16x16x4_f32(false, a, false, b0, (short)0, acc0, false, false);
      acc1 = __builtin_amdgcn_wmma_f32_16x16x4_f32(false, a, false, b1, (short)0, acc1, false, false);
      acc2 = __builtin_amdgcn_wmma_f32_16x16x4_f32(false, a, false, b2, (short)0, acc2, false, false);
      acc3 = __builtin_amdgcn_wmma_f32_16x16x4_f32(false, a, false, b3, (short)0, acc3, false, false);
    }
  }
#pragma unroll
  for (int r = 0; r < 8; ++r) {                // C/D: M = vgpr + 8*half
    float* Crow = C + (size_t)(tr + r + (half << 3)) * Nc + tc0 + lr;
    Crow[0]  = acc0[r];
    Crow[16] = acc1[r];
    Crow[32] = acc2[r];
    Crow[48] = acc3[r];
  }
}

// ---------------- LSTM gates GEMM (WMMA): gates[G,2048] = q*@W_ih^T + h@W_hh^T + b ----------
__global__ void wmma_gates(const float* __restrict__ Q,   // [G, 2*DOUT]
                           const float* __restrict__ Wih, // [4*DOUT, 2*DOUT] row-major
                           const float* __restrict__ Hs,  // [G, DOUT]
                           const float* __restrict__ Whh, // [4*DOUT, DOUT]
                           const float* __restrict__ bih, const float* __restrict__ bhh,
                           float* __restrict__ gates) {   // [G, 4*DOUT]
  int wave = blockIdx.x * 8 + (threadIdx.x >> 5);
  int lane = threadIdx.x & 31;
  int half = lane >> 4;
  int lr   = lane & 15;
  const int tilesPerRow = (4 * DOUT) >> 4;   // 128
  int tr = (wave / tilesPerRow) << 4;
  int tc = (wave % tilesPerRow) << 4;

  v8f acc = {};
  {  // segment 1: K = 2*DOUT over (Q, W_ih^T)  -> B[k][j] = Wih[j*2*DOUT + k]
    const float* Arow = Q   + (size_t)(tr + lr) * (2 * DOUT);
    const float* Bcol = Wih + (size_t)(tc + lr) * (2 * DOUT);
    for (int k0 = 0; k0 < 2 * DOUT; k0 += 4) {
      int ka = k0 + (half << 1);
      v2f a; a.x = Arow[ka]; a.y = Arow[ka + 1];
      v2f b; b.x = Bcol[ka]; b.y = Bcol[ka + 1];
      acc = __builtin_amdgcn_wmma_f32_16x16x4_f32(false, a, false, b, (short)0, acc, false, false);
    }
  }
  {  // segment 2: K = DOUT over (Hs, W_hh^T)
    const float* Arow = Hs  + (size_t)(tr + lr) * DOUT;
    const float* Bcol = Whh + (size_t)(tc + lr) * DOUT;
    for (int k0 = 0; k0 < DOUT; k0 += 4) {
      int ka = k0 + (half << 1);
      v2f a; a.x = Arow[ka]; a.y = Arow[ka + 1];
      v2f b; b.x = Bcol[ka]; b.y = Bcol[ka + 1];
      acc = __builtin_amdgcn_wmma_f32_16x16x4_f32(false, a, false, b, (short)0, acc, false, false);
    }
  }
  float bias = bih[tc + lr] + bhh[tc + lr];
#pragma unroll
  for (int r = 0; r < 8; ++r)
    gates[(size_t)(tr + r + (half << 3)) * (4 * DOUT) + tc + lr] = acc[r] + bias;
}

// ---------------- edge scatter: agg[dst] += H[src] * norm (one wave per edge) ----------------
__global__ void edge_agg(const long long* __restrict__ src, const long long* __restrict__ dst,
                         const float* __restrict__ nrm, const float* __restrict__ H,
                         float* __restrict__ agg) {
  int e    = blockIdx.x * 8 + (threadIdx.x >> 5);
  int lane = threadIdx.x & 31;
  int s = (int)src[e];
  int d = (int)dst[e];
  float w = nrm[e];
  const float* hv = H + (size_t)s * DH;
  float* av = agg + (size_t)d * DH;
#pragma unroll
  for (int f0 = 0; f0 < DH; f0 += 128) {
    int f = f0 + (lane << 2);
    float4 h4 = *(const float4*)&hv[f];
    atomicAdd(&av[f + 0], h4.x * w);
    atomicAdd(&av[f + 1], h4.y * w);
    atomicAdd(&av[f + 2], h4.z * w);
    atomicAdd(&av[f + 3], h4.w * w);
  }
}

// ---------------- GCN epilogue: out = [relu](agg + H*dinv^2 + b) ----------------
__global__ void gcn_epilogue(const float* __restrict__ agg, const float* __restrict__ H,
                             const float* __restrict__ dinv, const float* __restrict__ bias,
                             float* __restrict__ out, int do_relu) {
  int idx = blockIdx.x * blockDim.x + threadIdx.x;   // N*DH exact
  int n = idx >> 9;           // /512
  int f = idx & 511;
  float di = dinv[n];
  float v = agg[idx] + H[idx] * di * di + bias[f];
  out[idx] = do_relu ? fmaxf(v, 0.0f) : v;
}

// ---------------- LSTM cell elementwise ----------------
__global__ void lstm_cell(const float* __restrict__ gates, float* __restrict__ hs,
                          float* __restrict__ cs) {
  int idx = blockIdx.x * blockDim.x + threadIdx.x;   // G*DOUT exact
  int g = idx >> 9;
  int c = idx & 511;
  const float* gr = gates + ((size_t)g << 11);       // g*2048
  float gi = gr[c];
  float gf = gr[512 + c];
  float gc = gr[1024 + c];
  float go = gr[1536 + c];
  float cn = sigmoidf_(gf) * cs[idx] + sigmoidf_(gi) * tanhf(gc);
  cs[idx] = cn;
  hs[idx] = sigmoidf_(go) * tanhf(cn);
}

// ---------------- attention: e[n] = <h2[n], hs[batch[n]]>, segment max ----------------
__global__ void attn_dot(const float* __restrict__ h2, const long long* __restrict__ batch,
                         const float* __restrict__ hs, float* __restrict__ e,
                         unsigned* __restrict__ emax_enc) {
  int n    = blockIdx.x * 8 + (threadIdx.x >> 5);    // one wave per node
  int lane = threadIdx.x & 31;
  int g = (int)batch[n];
  const float* hv = h2 + ((size_t)n << 9);
  const float* qv = hs + ((size_t)g << 9);
  float s = 0.0f;
#pragma unroll
  for (int f0 = 0; f0 < DOUT; f0 += 128) {
    int f = f0 + (lane << 2);
    float4 a4 = *(const float4*)&hv[f];
    float4 b4 = *(const float4*)&qv[f];
    s += a4.x * b4.x + a4.y * b4.y + a4.z * b4.z + a4.w * b4.w;
  }
#pragma unroll
  for (int off = 16; off > 0; off >>= 1) s += __shfl_xor(s, off, 32);
  if (lane == 0) {
    e[n] = s;
    atomicMax(&emax_enc[g], flt_flip(s));
  }
}

__global__ void attn_exp(float* __restrict__ e, const long long* __restrict__ batch,
                         const unsigned* __restrict__ emax_enc, float* __restrict__ denom) {
  int n = blockIdx.x * blockDim.x + threadIdx.x;     // N exact
  int g = (int)batch[n];
  unsigned u = emax_enc[g];
  float m = (u == 0u) ? 0.0f : flt_unflip(u);        // empty segment -> 0 (matches isfinite guard)
  float ex = expf(e[n] - m);
  e[n] = ex;                                         // overwrite e with exp
  atomicAdd(&denom[g], ex);
}

// one wave per node: r[g] += a_n * h2[n]
__global__ void attn_wsum(const float* __restrict__ ex, const float* __restrict__ denom,
                          const long long* __restrict__ batch, const float* __restrict__ h2,
                          float* __restrict__ r) {
  int n    = blockIdx.x * 8 + (threadIdx.x >> 5);
  int lane = threadIdx.x & 31;
  int g = (int)batch[n];
  float a = ex[n] / fmaxf(denom[g], 1e-12f);
  const float* hv = h2 + ((size_t)n << 9);
  float* rv = r + ((size_t)g << 9);
#pragma unroll
  for (int f0 = 0; f0 < DOUT; f0 += 128) {
    int f = f0 + (lane << 2);
    float4 h4 = *(const float4*)&hv[f];
    atomicAdd(&rv[f + 0], a * h4.x);
    atomicAdd(&rv[f + 1], a * h4.y);
    atomicAdd(&rv[f + 2], a * h4.z);
    atomicAdd(&rv[f + 3], a * h4.w);
  }
}

__global__ void update_qstar(const float* __restrict__ hs, const float* __restrict__ r,
                             float* __restrict__ q) {
  int idx = blockIdx.x * blockDim.x + threadIdx.x;   // G*2*DOUT exact
  int g = idx >> 10;
  int j = idx & 1023;
  q[idx] = (j < DOUT) ? hs[(g << 9) + j] : r[(g << 9) + (j - DOUT)];
}

// ---------------- launch ----------------
extern "C" void kernel_launch(void* const* d_in, const int* in_sizes, int n_in,
                              void* d_out, int out_size, void* d_ws, size_t ws_size,
                              hipStream_t stream) {
  (void)in_sizes; (void)n_in; (void)out_size; (void)ws_size;

  const float*     x    = (const float*)d_in[0];
  const long long* ei   = (const long long*)d_in[1];   // [2, E] int64
  const long long* bat  = (const long long*)d_in[2];   // [N] int64 (sorted)
  const float*     W1   = (const float*)d_in[3];
  const float*     b1   = (const float*)d_in[4];
  const float*     W2   = (const float*)d_in[5];
  const float*     b2   = (const float*)d_in[6];
  const float*     W_ih = (const float*)d_in[7];
  const float*     W_hh = (const float*)d_in[8];
  const float*     b_ih = (const float*)d_in[9];
  const float*     b_hh = (const float*)d_in[10];
  const long long* src  = ei;
  const long long* dst  = ei + EE;

  float* q_star = (float*)d_out;                      // [G, 2*DOUT] is the output itself

  // workspace carve-up
  float* ws = (float*)d_ws;
  float* H    = ws;                      ws += (size_t)NN * DH;   // GEMM output (H1, then H2)
  float* AGG  = ws;                      ws += (size_t)NN * DH;   // edge aggregation
  float* HF   = ws;                      ws += (size_t)NN * DH;   // h (layer1 out), then h2
  float* DINV = ws;                      ws += NN;                // degree -> rsqrt in place
  float* NORM = ws;                      ws += EE;                // per-edge norm
  float* EBUF = ws;                      ws += NN;                // e, then exp(e-max)
  float* HS   = ws;                      ws += (size_t)GG * DOUT;
  float* CS   = ws;                      ws += (size_t)GG * DOUT;
  float* RB   = ws;                      ws += (size_t)GG * DOUT;
  float* GT   = ws;                      ws += (size_t)GG * 4 * DOUT;
  unsigned* EMX = (unsigned*)ws;         ws += GG;
  float* DEN  = ws;                      ws += GG;

  const int B  = 256;
  const int ND = NN * DH;                              // 16,777,216 (exact multiple of 256)
  const int gemmBlocks = (NN / 128) * (DH / 64);       // 256 * 8 = 2048

  // ---- GCN normalization ----
  set_ones<<<NN / B, B, 0, stream>>>(DINV, NN);
  deg_edges<<<EE / B, B, 0, stream>>>(dst, DINV, EE);
  rsqrt_inplace<<<NN / B, B, 0, stream>>>(DINV, NN);
  norm_kernel<<<EE / B, B, 0, stream>>>(src, dst, DINV, NORM, EE);

  // ---- layer 1: H = x@W1 ; agg ; h = relu(...) ----
  wmma_gemm_f32_lds<<<gemmBlocks, B, 0, stream>>>(x, W1, H, DIN, DH);
  zero_f32<<<ND / B, B, 0, stream>>>(AGG, ND);
  edge_agg<<<EE / 8, B, 0, stream>>>(src, dst, NORM, H, AGG);
  gcn_epilogue<<<ND / B, B, 0, stream>>>(AGG, H, DINV, b1, HF, 1);

  // ---- layer 2: H = h@W2 ; agg ; h2 = ... (into HF) ----
  wmma_gemm_f32_lds<<<gemmBlocks, B, 0, stream>>>(HF, W2, H, DH, DOUT);
  zero_f32<<<ND / B, B, 0, stream>>>(AGG, ND);
  edge_agg<<<EE / 8, B, 0, stream>>>(src, dst, NORM, H, AGG);
  gcn_epilogue<<<ND / B, B, 0, stream>>>(AGG, H, DINV, b2, HF, 0);   // HF = h2

  // ---- Set2Set ----
  zero_f32<<<GG * DOUT / B, B, 0, stream>>>(HS, GG * DOUT);
  zero_f32<<<GG * DOUT / B, B, 0, stream>>>(CS, GG * DOUT);
  zero_f32<<<GG * 2 * DOUT / B, B, 0, stream>>>(q_star, GG * 2 * DOUT);

  const int gateBlocks = (GG / 16) * ((4 * DOUT) / 16) / 8;          // 256
  for (int step = 0; step < STEPS; ++step) {
    wmma_gates<<<gateBlocks, B, 0, stream>>>(q_star, W_ih, HS, W_hh, b_ih, b_hh, GT);
    lstm_cell<<<GG * DOUT / B, B, 0, stream>>>(GT, HS, CS);
    zero_f32<<<GG * DOUT / B, B, 0, stream>>>(RB, GG * DOUT);
    zero_small<<<1, B, 0, stream>>>(EMX, DEN, GG);
    attn_dot<<<NN / 8, B, 0, stream>>>(HF, bat, HS, EBUF, EMX);
    attn_exp<<<NN / B, B, 0, stream>>>(EBUF, bat, EMX, DEN);
    attn_wsum<<<NN / 8, B, 0, stream>>>(EBUF, DEN, bat, HF, RB);
    update_qstar<<<GG * 2 * DOUT / B, B, 0, stream>>>(HS, RB, q_star);
  }
}